// Anki_42820823941684
// MI455X (gfx1250) — compile-verified
//
#include <hip/hip_runtime.h>
#include <stdint.h>

// ---------------------------------------------------------------------------
// Anki scheduler scan: SEQ=512 serial steps over BATCH=65536 independent
// sequences. Memory-bound stream (~512 MB @ 23.3 TB/s => ~22us floor).
// Strategy: per-lane 2 batch elements (float4 accesses), TRIPLE-buffered
// async global->LDS prefetch (CDNA5 GLOBAL_LOAD_ASYNC_TO_LDS_B128, ASYNCcnt,
// 32 wave-wide 512B loads in flight => ~16 MB chip-wide MLP), non-temporal
// 128-bit output stores so the 256 MB write-once stream skips L2 residency.
// ---------------------------------------------------------------------------

#define SEQ   512
#define BATCH 65536
#define TILE  16                  // timesteps staged per LDS buffer
#define NTILES (SEQ / TILE)       // 32
#define NBUF  3                   // ring depth: 2 tiles always in flight
#define WAVES_PER_BLOCK 4
#define THREADS (WAVES_PER_BLOCK * 32)

typedef float f4  __attribute__((ext_vector_type(4)));
typedef int   v4i __attribute__((vector_size(16)));   // matches builtin param type

#define S_MAXV 36500.0f
#define S_MINV 0.01f

__device__ __forceinline__ void anki_step(
    float dt, float r, float& ivl, float& ease,
    float w0, float w1, float w2, float w3, float w4, float w5, float w6)
{
    const bool isf = (ivl == 0.0f) || (ease == 0.0f);

    float ne = ease;
    ne = (r == 1.0f) ? (ease - 0.20f) : ne;
    ne = (r == 2.0f) ? (ease - 0.15f) : ne;
    ne = (r == 4.0f) ? (ease + 0.15f) : ne;
    ne = fminf(fmaxf(ne, 1.3f), 5.5f);

    const float dl   = dt - ivl;
    const bool  pass = (r > 1.0f);
    const bool  early     = pass && (dl <  0.0f);
    const bool  non_early = pass && (dl >= 0.0f);

    const float elapsed = ivl + dl;
    const float e_hard  = fmaxf(elapsed * w4, (ivl * w4) * 0.5f);
    const float e_good  = ivl * ne;
    const float e_easy  = e_good * w3;
    const float ivl_e   = (r == 2.0f) ? e_hard : ((r == 4.0f) ? e_easy : e_good);

    const float n_hard  = ivl * w4;
    const float n_good  = (ivl + dl * 0.5f)  * ne;
    const float n_easy  = ((ivl + dl * 0.25f) * ne) * w3;
    const float ivl_ne  = (r == 2.0f) ? n_hard : ((r == 4.0f) ? n_easy : n_good);

    const float calc = (early ? ivl_e : (non_early ? ivl_ne : 0.0f)) * w6;

    float ni = pass ? calc : ivl;
    ni = (r == 1.0f)              ? (ivl * w5) : ni;
    ni = (isf && (r <  4.0f))     ? w0         : ni;
    ni = (isf && (r == 4.0f))     ? w1         : ni;
    ne = isf                      ? w2         : ne;
    ni = fminf(fmaxf(ni, S_MINV), S_MAXV);

    ivl  = ni;
    ease = ne;
}

__global__ __launch_bounds__(THREADS, 1)
void Anki_scan_kernel(const float* __restrict__ in,
                      const float* __restrict__ w,
                      float* __restrict__ out)
{
    // Per-wave private staging ring: [buf][wave][step][lane], 16B per lane.
    // 3 * 4 * 16 * 32 * 16B = 96 KB per block.
    __shared__ f4 stage[NBUF][WAVES_PER_BLOCK][TILE][32];

    const int tid  = threadIdx.x;
    const int wave = tid >> 5;
    const int lane = tid & 31;

    // Uniform weights -> scalar loads
    const float w0 = w[0], w1 = w[1], w2 = w[2], w3 = w[3];
    const float w4 = w[4], w5 = w[5], w6 = w[6];

    // This lane owns batch elements b0 and b0+1.
    const int    b0     = blockIdx.x * (THREADS * 2) + wave * 64 + lane * 2;
    const size_t f4base = (size_t)(b0 >> 1);      // float4 column index in a row
    const f4* in4  = (const f4*)in;
    f4*       out4 = (f4*)out;

    float ivl0 = 0.0f, ease0 = 0.0f, ivl1 = 0.0f, ease1 = 0.0f;

    // Prime buffers 0 and 1 with tiles 0 and 1 (32 async loads in flight).
    #pragma unroll
    for (int p = 0; p < 2; ++p) {
        #pragma unroll
        for (int s = 0; s < TILE; ++s) {
            const f4* g = in4 + ((size_t)(p * TILE + s) * (BATCH / 2) + f4base);
            __builtin_amdgcn_global_load_async_to_lds_b128(
                (v4i*)g, (v4i*)&stage[p][wave][s][lane], 0, 0);
        }
    }

    int buf  = 0;                 // buffer holding current tile
    int bufi = 2;                 // buffer to fill with tile+2

    for (int tile = 0; tile < NTILES; ++tile) {
        if (tile + 2 < NTILES) {
            // Keep 2 tiles ahead in flight: issue tile+2, then wait until only
            // those 32 (tiles t+1, t+2) remain outstanding => tile t is ready.
            const int t0 = (tile + 2) * TILE;
            #pragma unroll
            for (int s = 0; s < TILE; ++s) {
                const f4* g = in4 + ((size_t)(t0 + s) * (BATCH / 2) + f4base);
                __builtin_amdgcn_global_load_async_to_lds_b128(
                    (v4i*)g, (v4i*)&stage[bufi][wave][s][lane], 0, 0);
            }
            asm volatile("s_wait_asynccnt 0x20" ::: "memory");
        } else if (tile + 1 < NTILES) {
            asm volatile("s_wait_asynccnt 0x10" ::: "memory");
        } else {
            asm volatile("s_wait_asynccnt 0x0" ::: "memory");
        }

        #pragma unroll
        for (int s = 0; s < TILE; ++s) {
            const int t = tile * TILE + s;
            const f4 x = stage[buf][wave][s][lane];   // {dt0, r0, dt1, r1}

            anki_step(x.x, x.y, ivl0, ease0, w0, w1, w2, w3, w4, w5, w6);
            anki_step(x.z, x.w, ivl1, ease1, w0, w1, w2, w3, w4, w5, w6);

            f4 o;
            o.x = ivl0; o.y = ease0; o.z = ivl1; o.w = ease1;
            __builtin_nontemporal_store(o, out4 + ((size_t)t * (BATCH / 2) + f4base));
        }

        // advance ring
        buf  = (buf  == NBUF - 1) ? 0 : buf  + 1;
        bufi = (bufi == NBUF - 1) ? 0 : bufi + 1;
    }

    // final_state tail: d_out[SEQ*BATCH*2 + b*2 + c]
    f4 fs;
    fs.x = ivl0; fs.y = ease0; fs.z = ivl1; fs.w = ease1;
    __builtin_nontemporal_store(fs, out4 + ((size_t)SEQ * (BATCH / 2) + f4base));
}

extern "C" void kernel_launch(void* const* d_in, const int* in_sizes, int n_in,
                              void* d_out, int out_size, void* d_ws, size_t ws_size,
                              hipStream_t stream)
{
    const float* in = (const float*)d_in[0];   // (512, 65536, 2) f32
    const float* w  = (const float*)d_in[1];   // (7,) f32
    float* out = (float*)d_out;                // outputs (512,65536,2) ++ final (65536,2)

    const int grid = BATCH / (THREADS * 2);    // 256 blocks
    Anki_scan_kernel<<<grid, THREADS, 0, stream>>>(in, w, out);
}